// GCN_47287589929768
// MI455X (gfx1250) — compile-verified
//
#include <hip/hip_runtime.h>
#include <math.h>

typedef __attribute__((ext_vector_type(2))) float v2f;
typedef __attribute__((ext_vector_type(4))) float v4f;
typedef __attribute__((ext_vector_type(8))) float v8f;
typedef __attribute__((ext_vector_type(4))) int   v4i;

// Native GLOBAL_ATOMIC_ADD_F32 (no-return). Plain atomicAdd(float*) may lower
// to a CAS loop; unsafeAtomicAdd maps to the hardware FP atomic (RNE, fixed
// denorm behavior per ISA ch.12), which is what we want for L2-resident agg.
__device__ __forceinline__ void atomAddF(float* p, float v) {
    (void)unsafeAtomicAdd(p, v);
}

// ---------------------------------------------------------------------------
// Degree / dinv (computed once, reused by all three layers)
// ---------------------------------------------------------------------------
__global__ void k_cnt_init(unsigned int* __restrict__ cnt, int N) {
    int i = blockIdx.x * blockDim.x + threadIdx.x;
    if (i < N) cnt[i] = 0u;
}

__global__ void k_deg_count(const int* __restrict__ dst, unsigned int* __restrict__ cnt, int E) {
    const int tid = blockIdx.x * blockDim.x + threadIdx.x;
    const int nt  = gridDim.x * blockDim.x;
    const int E4  = E >> 2;
    const v4i* __restrict__ d4p = (const v4i*)dst;
    for (int q = tid; q < E4; q += nt) {
        v4i d = __builtin_nontemporal_load(d4p + q);   // b128 NT stream
        atomicAdd(&cnt[d.x], 1u);
        atomicAdd(&cnt[d.y], 1u);
        atomicAdd(&cnt[d.z], 1u);
        atomicAdd(&cnt[d.w], 1u);
    }
    for (int e = (E & ~3) + tid; e < E; e += nt) atomicAdd(&cnt[dst[e]], 1u);
}

__global__ void k_dinv(const unsigned int* __restrict__ cnt, float* __restrict__ dinv, int N) {
    int i = blockIdx.x * blockDim.x + threadIdx.x;
    if (i < N) dinv[i] = rsqrtf((float)cnt[i] + 1.0f);  // +1 self-loop, deg >= 1
}

// ---------------------------------------------------------------------------
// Fused transform + self-loop init via V_WMMA_F32_16X16X4_F32.
//   A = W^T zero-padded to 16x4 (M = out-feature, K = in-feature) -- invariant
//   B = 16 node rows as 4x16   (K = in-feature, N = node)
//   D[m][n] = xw[node n][m]
// Per-lane (half = lane/16, mn = lane%16, reg r):
//   A reg r : A[mn][2*half + r]      B reg r : B[2*half + r][mn]
//   D reg v : D[v + 8*half][mn]  =>  half==0 lane mn holds ALL outputs of
//                                    node base+mn in c[0..OUT-1]
// Writes xw[row*4 + j] and agg[row*4 + j] = xw * dinv[row]^2 (self-loop term)
// with one coalesced vector store each. Full 16-node tiles only.
// ---------------------------------------------------------------------------
template <int IN, int OUT>
__global__ void k_transform_wmma(const float* __restrict__ h, const float* __restrict__ W,
                                 const float* __restrict__ dinv,
                                 float* __restrict__ xw, float* __restrict__ agg,
                                 int ntiles) {
    const int lane   = threadIdx.x & 31;
    const int wave   = (blockIdx.x * blockDim.x + threadIdx.x) >> 5;
    const int nwaves = (gridDim.x * blockDim.x) >> 5;
    const int half   = lane >> 4;
    const int mn     = lane & 15;
    const int k0     = 2 * half;
    const int k1     = k0 + 1;

    // A operand (weights): loaded once, lane-predicated loads hoisted out of loop
    v2f a;
    a.x = (mn < OUT && k0 < IN) ? W[k0 * OUT + mn] : 0.0f;
    a.y = (mn < OUT && k1 < IN) ? W[k1 * OUT + mn] : 0.0f;

    for (int t = wave; t < ntiles; t += nwaves) {
        const int row = (t << 4) + mn;
        v2f b;
        if (IN == 2) {
            // stride-2 input; unconditional load, select zeros for K>=2 half
            const float2 hv = *reinterpret_cast<const float2*>(h + (size_t)row * 2);
            b.x = half ? 0.0f : hv.x;
            b.y = half ? 0.0f : hv.y;
        } else {  // IN == 4, stride-4 input
            const float2 hv = *reinterpret_cast<const float2*>(h + (size_t)row * 4 + k0);
            b.x = hv.x;
            b.y = hv.y;
        }

        v8f c = {};
        c = __builtin_amdgcn_wmma_f32_16x16x4_f32(
                /*neg_a=*/false, a, /*neg_b=*/false, b,
                /*c_mod=*/(short)0, c, /*reuse_a=*/false, /*reuse_b=*/false);

        if (half == 0) {  // lanes 0..15 own node base+mn, outputs in c[0..OUT-1]
            float w = dinv[row];
            w = w * w;
            if (OUT == 4) {
                v4f o = {c[0], c[1], c[2], c[3]};
                *reinterpret_cast<v4f*>(xw + (size_t)row * 4) = o;
                v4f s = {o.x * w, o.y * w, o.z * w, o.w * w};
                *reinterpret_cast<v4f*>(agg + (size_t)row * 4) = s;
            } else {  // OUT == 2
                v2f o = {c[0], c[1]};
                *reinterpret_cast<v2f*>(xw + (size_t)row * 4) = o;
                v2f s = {o.x * w, o.y * w};
                *reinterpret_cast<v2f*>(agg + (size_t)row * 4) = s;
            }
        }
    }
}

// Scalar tail for N % 16 != 0 (never runs for N = 500000; kept for generality)
template <int IN, int OUT>
__global__ void k_transform_tail(const float* __restrict__ h, const float* __restrict__ W,
                                 const float* __restrict__ dinv,
                                 float* __restrict__ xw, float* __restrict__ agg,
                                 int start, int N) {
    int i = start + blockIdx.x * blockDim.x + threadIdx.x;
    if (i < N) {
        const int hs = (IN == 2) ? 2 : 4;
        float w = dinv[i];
        w = w * w;
#pragma unroll
        for (int j = 0; j < OUT; ++j) {
            float acc = 0.0f;
#pragma unroll
            for (int k = 0; k < IN; ++k) acc += h[(size_t)i * hs + k] * W[k * OUT + j];
            xw[(size_t)i * 4 + j]  = acc;
            agg[(size_t)i * 4 + j] = acc * w;
        }
    }
}

// ---------------------------------------------------------------------------
// Edge scatter: agg[dst] += xw[src] * dinv[src]*dinv[dst]
// int4 NT streams for src/dst; node arrays (dinv/xw/agg) stay hot in L2.
// ---------------------------------------------------------------------------
template <int OUT>
__device__ __forceinline__ void edge_one(int s, int d,
                                         const float* __restrict__ dinv,
                                         const float* __restrict__ xw,
                                         float* __restrict__ agg) {
    float nrm = dinv[s] * dinv[d];
    if (OUT == 4) {
        v4f v = *reinterpret_cast<const v4f*>(xw + (size_t)s * 4);
        atomAddF(&agg[(size_t)d * 4 + 0], v.x * nrm);
        atomAddF(&agg[(size_t)d * 4 + 1], v.y * nrm);
        atomAddF(&agg[(size_t)d * 4 + 2], v.z * nrm);
        atomAddF(&agg[(size_t)d * 4 + 3], v.w * nrm);
    } else {
        v2f v = *reinterpret_cast<const v2f*>(xw + (size_t)s * 4);
        atomAddF(&agg[(size_t)d * 4 + 0], v.x * nrm);
        atomAddF(&agg[(size_t)d * 4 + 1], v.y * nrm);
    }
}

template <int OUT>
__global__ void k_edge(const int* __restrict__ src, const int* __restrict__ dst,
                       const float* __restrict__ dinv, const float* __restrict__ xw,
                       float* __restrict__ agg, int E) {
    const int tid = blockIdx.x * blockDim.x + threadIdx.x;
    const int nt  = gridDim.x * blockDim.x;
    const int E4  = E >> 2;
    const v4i* __restrict__ s4p = (const v4i*)src;
    const v4i* __restrict__ d4p = (const v4i*)dst;
    for (int q = tid; q < E4; q += nt) {
        v4i s = __builtin_nontemporal_load(s4p + q);   // b128 NT stream
        v4i d = __builtin_nontemporal_load(d4p + q);
        edge_one<OUT>(s.x, d.x, dinv, xw, agg);
        edge_one<OUT>(s.y, d.y, dinv, xw, agg);
        edge_one<OUT>(s.z, d.z, dinv, xw, agg);
        edge_one<OUT>(s.w, d.w, dinv, xw, agg);
    }
    for (int e = (E & ~3) + tid; e < E; e += nt)
        edge_one<OUT>(src[e], dst[e], dinv, xw, agg);
}

// ---------------------------------------------------------------------------
// out = (tanh?)(agg + b); OUT==4 -> stride-4 hbuf, OUT==2 -> stride-2 d_out
// ---------------------------------------------------------------------------
template <int OUT, bool TANH>
__global__ void k_finalize(const float* __restrict__ agg, const float* __restrict__ bias,
                           float* __restrict__ out, int N) {
    int i = blockIdx.x * blockDim.x + threadIdx.x;
    if (i < N) {
        if (OUT == 4) {
            v4f v = *reinterpret_cast<const v4f*>(agg + (size_t)i * 4);
            v.x += bias[0]; v.y += bias[1]; v.z += bias[2]; v.w += bias[3];
            if (TANH) { v.x = tanhf(v.x); v.y = tanhf(v.y); v.z = tanhf(v.z); v.w = tanhf(v.w); }
            *reinterpret_cast<v4f*>(out + (size_t)i * 4) = v;
        } else {
            v2f v = *reinterpret_cast<const v2f*>(agg + (size_t)i * 4);
            v.x += bias[0]; v.y += bias[1];
            if (TANH) { v.x = tanhf(v.x); v.y = tanhf(v.y); }
            *reinterpret_cast<v2f*>(out + (size_t)i * 2) = v;
        }
    }
}

// ---------------------------------------------------------------------------
extern "C" void kernel_launch(void* const* d_in, const int* in_sizes, int n_in,
                              void* d_out, int out_size, void* d_ws, size_t ws_size,
                              hipStream_t stream) {
    const float* x  = (const float*)d_in[0];
    const int*   ei = (const int*)d_in[1];
    const float* W1 = (const float*)d_in[2];
    const float* b1 = (const float*)d_in[3];
    const float* W2 = (const float*)d_in[4];
    const float* b2 = (const float*)d_in[5];
    const float* W3 = (const float*)d_in[6];
    const float* b3 = (const float*)d_in[7];

    const int N = in_sizes[0] / 2;   // x is [N, 2]
    const int E = in_sizes[1] / 2;   // edge_index is [2, E]
    const int* src = ei;
    const int* dst = ei + E;

    // 16B-aligned workspace slices (unit rounded to multiple of 4 floats)
    const size_t NU = ((size_t)N + 3) & ~(size_t)3;
    float* ws   = (float*)d_ws;
    float*        dinv = ws;                        // N floats
    float*        xw   = ws + NU;                   // 4N floats (stride-4 padded)
    float*        agg  = ws + 5 * NU;               // 4N floats
    float*        hbuf = ws + 9 * NU;               // 4N floats
    unsigned int* cnt  = (unsigned int*)(ws + 13 * NU);  // N uints (degree counts)
    float* outp = (float*)d_out;

    const int TB = 256;
    const int NB = (N + TB - 1) / TB;
    const int EB = 2048;              // edge kernels: ~0.5M threads over 4M int4 groups
    const int WB = 512;               // WMMA transform: 4096 waves over 31250 tiles

    const int full   = N & ~15;
    const int ntiles = full >> 4;
    const bool tail  = (full < N);

    // --- degree / dinv once ---
    k_cnt_init<<<NB, TB, 0, stream>>>(cnt, N);
    k_deg_count<<<1024, TB, 0, stream>>>(dst, cnt, E);
    k_dinv<<<NB, TB, 0, stream>>>(cnt, dinv, N);

    // --- layer 1: in=2 out=4, tanh ---
    k_transform_wmma<2, 4><<<WB, TB, 0, stream>>>(x, W1, dinv, xw, agg, ntiles);
    if (tail) k_transform_tail<2, 4><<<1, 32, 0, stream>>>(x, W1, dinv, xw, agg, full, N);
    k_edge<4><<<EB, TB, 0, stream>>>(src, dst, dinv, xw, agg, E);
    k_finalize<4, true><<<NB, TB, 0, stream>>>(agg, b1, hbuf, N);

    // --- layer 2: in=4 out=4, tanh ---
    k_transform_wmma<4, 4><<<WB, TB, 0, stream>>>(hbuf, W2, dinv, xw, agg, ntiles);
    if (tail) k_transform_tail<4, 4><<<1, 32, 0, stream>>>(hbuf, W2, dinv, xw, agg, full, N);
    k_edge<4><<<EB, TB, 0, stream>>>(src, dst, dinv, xw, agg, E);
    k_finalize<4, true><<<NB, TB, 0, stream>>>(agg, b2, hbuf, N);

    // --- layer 3: in=4 out=2, no tanh, straight into d_out (stride 2) ---
    k_transform_wmma<4, 2><<<WB, TB, 0, stream>>>(hbuf, W3, dinv, xw, agg, ntiles);
    if (tail) k_transform_tail<4, 2><<<1, 32, 0, stream>>>(hbuf, W3, dinv, xw, agg, full, N);
    k_edge<2><<<EB, TB, 0, stream>>>(src, dst, dinv, xw, agg, E);
    k_finalize<2, false><<<NB, TB, 0, stream>>>(agg, b3, outp, N);
}